// CPUBouncingEmbedding_30399778521606
// MI455X (gfx1250) — compile-verified
//
#include <hip/hip_runtime.h>

// Embedding row gather: out[b,s,:] = weight[ids[b,s], :]
//   ids:    int32 [8, 4096]           (32768 rows)
//   weight: fp32  [128000, 1024]      (512 MB table, > L2)
//   out:    fp32  [8, 4096, 1024]     (128 MiB, streaming write-once)
//
// Pure bandwidth problem (~256 MiB traffic -> ~11 us at 23.3 TB/s).
// One 256-thread block processes 8 rows. Fast path (block fully in range,
// always true for 32768 rows): issue all 8 global_load_b128 gathers first
// (8 loads in flight per thread), then all 8 non-temporal b128 stores.
// Row indices are block-uniform -> scalar s_load for ids[].

typedef float v4f __attribute__((ext_vector_type(4)));

#define DIM            1024
#define VEC_PER_ROW    (DIM / 4)      // 256 x v4f per row
#define ROWS_PER_BLOCK 8
#define THREADS        VEC_PER_ROW    // 256 threads = 8 wave32

__global__ __launch_bounds__(THREADS) void
embed_gather_b128(const int* __restrict__ ids,
                  const v4f* __restrict__ weight,
                  v4f* __restrict__ out,
                  int n_rows)
{
    const int  tid  = threadIdx.x;                       // v4f column within row
    const long row0 = (long)blockIdx.x * ROWS_PER_BLOCK;

    if (row0 + ROWS_PER_BLOCK <= n_rows) {
        // ---- fast path: no per-row guards, maximize loads in flight ----
        v4f v[ROWS_PER_BLOCK];

#pragma unroll
        for (int r = 0; r < ROWS_PER_BLOCK; ++r) {
            const int id = ids[row0 + r];                // scalar s_load (uniform)
            v[r] = weight[(size_t)id * VEC_PER_ROW + (size_t)tid];  // global_load_b128
        }

#pragma unroll
        for (int r = 0; r < ROWS_PER_BLOCK; ++r) {
            __builtin_nontemporal_store(
                v[r], &out[(size_t)(row0 + r) * VEC_PER_ROW + (size_t)tid]);
        }
    } else {
        // ---- tail path (not taken for 32768 rows, kept for generality) ----
        for (int r = 0; r < ROWS_PER_BLOCK; ++r) {
            const long row = row0 + r;
            if (row >= n_rows) break;
            const int id = ids[row];
            const v4f v  = weight[(size_t)id * VEC_PER_ROW + (size_t)tid];
            __builtin_nontemporal_store(
                v, &out[(size_t)row * VEC_PER_ROW + (size_t)tid]);
        }
    }
}

extern "C" void kernel_launch(void* const* d_in, const int* in_sizes, int n_in,
                              void* d_out, int out_size, void* d_ws, size_t ws_size,
                              hipStream_t stream)
{
    // setup_inputs() order: input_ids (int32, B*S), weight (fp32, VOCAB*DIM)
    const int* ids    = (const int*)d_in[0];
    const v4f* weight = (const v4f*)d_in[1];
    v4f*       out    = (v4f*)d_out;

    const int n_rows = in_sizes[0];                      // 8 * 4096 = 32768
    const int grid   = (n_rows + ROWS_PER_BLOCK - 1) / ROWS_PER_BLOCK;

    embed_gather_b128<<<grid, THREADS, 0, stream>>>(ids, weight, out, n_rows);
}